// edge_conv_30219389895261
// MI455X (gfx1250) — compile-verified
//
#include <hip/hip_runtime.h>
#include <float.h>

#define BB    8
#define CC    64
#define NN    4096
#define CO    64
#define KNN   16
#define ROWS  64
#define CHUNK 128
#define NCH   (NN / CHUNK)
#define BPITCH 132   // floats per c-row of XBT: 16B-aligned rows, (4c+n)%64 banking

typedef float v2f __attribute__((ext_vector_type(2)));
typedef float v8f __attribute__((ext_vector_type(8)));

#if defined(__HIP_DEVICE_COMPILE__) && defined(__has_builtin)
#if __has_builtin(__builtin_amdgcn_global_load_async_to_lds_b128) && \
    __has_builtin(__builtin_amdgcn_s_wait_asynccnt)
#define HAVE_ASYNC 1
#endif
#endif

#if defined(HAVE_ASYNC)
// Builtin signature (from hipcc diagnostic): param0 = int32x4 AS1*, param1 = AS3 side.
typedef int v4i_vs __attribute__((vector_size(16)));
typedef __attribute__((address_space(1))) v4i_vs GV4;
typedef __attribute__((address_space(3))) v4i_vs LV4;
#define ASYNC_COPY_B128(gp, lp) \
    __builtin_amdgcn_global_load_async_to_lds_b128((GV4*)(gp), (LV4*)(lp), 0, 0)
#define WAIT_ASYNC() __builtin_amdgcn_s_wait_asynccnt(0)
#else
#define ASYNC_COPY_B128(gp, lp) \
    do { *(float4*)(lp) = *(const float4*)(gp); } while (0)
#define WAIT_ASYNC() do {} while (0)
#endif

// ---------------------------------------------------------------------------
// Kernel A: per (b, 64-point tile):
//   P[b,n,o] = sum_c x[b,c,n] * W1[o,c]                  (no bias)
//   R[b,n,o] = sum_c x[b,c,n] * W2[o,c] + b1[o] + b2[o] - P[b,n,o]
//   SQ[b,n]  = sum_c x[b,c,n]^2
// ---------------------------------------------------------------------------
__global__ __launch_bounds__(256) void edgeconv_precompute(
    const float* __restrict__ x,  const float* __restrict__ W1,
    const float* __restrict__ b1, const float* __restrict__ W2,
    const float* __restrict__ b2, float* __restrict__ P,
    float* __restrict__ R,        float* __restrict__ SQ)
{
    __shared__ float xts[64][65];          // xts[j][c] = xt[n0+j][c], padded
    const int b   = blockIdx.x >> 6;       // N/64 = 64 tiles per batch
    const int n0  = (blockIdx.x & 63) * 64;
    const int tid = threadIdx.x;

    for (int i = tid; i < 64 * 64; i += 256) {
        int c = i >> 6, j = i & 63;        // coalesced: 64 consecutive n per c
        xts[j][c] = x[(b * CC + c) * NN + n0 + j];
    }
    __syncthreads();

    if (tid < 64) {
        float s = 0.f;
        for (int c = 0; c < CC; ++c) { float v = xts[tid][c]; s += v * v; }
        SQ[b * NN + n0 + tid] = s;
    }

    const int j = tid & 63;
    for (int oo = tid >> 6; oo < CO; oo += 4) {
        float p = 0.f, q = 0.f;
        for (int c = 0; c < CC; ++c) {
            float v = xts[j][c];
            p += v * W1[oo * CC + c];
            q += v * W2[oo * CC + c];
        }
        int base = (b * NN + n0 + j) * CO + oo;
        P[base] = p;
        R[base] = q + b1[oo] + b2[oo] - p;
    }
}

// ---------------------------------------------------------------------------
// Kernel B: one block = (batch b, 64 consecutive rows), 8 waves.
// Phase 1: double-buffered async global->LDS column chunks overlapped with
//          2x2-register-tiled WMMA f32 16x16x4 Gram + per-row top-16 select.
// Phase 2: y[n,o] = relu(max_k (R[n,o] + P[nbr_k, o])), coalesced store.
// ---------------------------------------------------------------------------
__global__ __launch_bounds__(256) void edgeconv_knn_aggregate(
    const float* __restrict__ x,  const float* __restrict__ P,
    const float* __restrict__ R,  const float* __restrict__ SQ,
    float* __restrict__ out)
{
    __shared__ float XA[ROWS][65];                       // row tile, (m+c)%64 banks
    __shared__ __align__(16) float XBT[2][CC][BPITCH];   // c-major dbl-buffered cols
    __shared__ float ST[ROWS][CHUNK + 1];                // 129%64==1: clean scans
    __shared__ float sel_d[ROWS][KNN];                   // sorted ascending dists
    __shared__ int   sel_i[ROWS][KNN];                   // neighbor indices
    __shared__ float sqrow[ROWS];
    __shared__ float sqc[2][CHUNK];

    const int b    = blockIdx.x >> 6;        // N/ROWS = 64 blocks per batch
    const int row0 = (blockIdx.x & 63) * ROWS;
    const int tid  = threadIdx.x;
    const int lane = tid & 31;
    const int wave = tid >> 5;

    // ---- prologue: prefetch chunk 0 (async), load row tile (regular) ----
    {
        const float* gx = x + (size_t)(b * CC) * NN;     // + c*NN + n
#pragma unroll
        for (int t = 0; t < 8; ++t) {                    // 2048 x 16B transfers
            int idx = tid + t * 256;
            int c   = idx >> 5;
            int j16 = (idx & 31) << 2;                   // 4 floats per transfer
            ASYNC_COPY_B128(gx + c * NN + j16, &XBT[0][c][j16]);
        }
    }
    for (int i = tid; i < ROWS * CC; i += 256) {
        int c = i >> 6, j = i & 63;                      // coalesced 64-float runs
        XA[j][c] = x[(b * CC + c) * NN + row0 + j];
    }
    if (tid < CHUNK) sqc[0][tid] = SQ[b * NN + tid];
    if (tid < ROWS) {
        sqrow[tid] = SQ[b * NN + row0 + tid];
        for (int j = 0; j < KNN; ++j) { sel_d[tid][j] = FLT_MAX; sel_i[tid][j] = 0; }
    }

    // 8 waves, each owns a 2x2 block of 16x16 tiles:
    //   row tiles {2*rtp, 2*rtp+1} (rtp = wave>>2) -> 4 row tiles = 64 rows
    //   col tiles {2*ctp, 2*ctp+1} (ctp = wave&3)  -> 8 col tiles = 128 cols
    const int rtp  = wave >> 2;
    const int ctp  = wave & 3;
    const int l16  = lane & 15;
    const int ch   = 2 * (lane >> 4);        // 32-bit A/B layout: lanes16-31 = K 2,3
    const int m0   = (2 * rtp) * 16 + l16;   // A fragment rows
    const int m1   = m0 + 16;
    const int n0c  = (2 * ctp) * 16 + l16;   // B fragment cols
    const int n1c  = n0c + 16;

    for (int ci = 0; ci < NCH; ++ci) {
        const int cur = ci & 1, nxt = cur ^ 1;
        const int c0  = ci * CHUNK;

        // current buffer becomes visible to every wave here; all waves have
        // finished reading the buffer we are about to overwrite (it was last
        // read before the *previous* iteration's post-compute barrier).
        WAIT_ASYNC();
        __syncthreads();

        if (ci + 1 < NCH) {                  // prefetch chunk ci+1 into nxt
            const float* gx = x + (size_t)(b * CC) * NN + (c0 + CHUNK);
#pragma unroll
            for (int t = 0; t < 8; ++t) {
                int idx = tid + t * 256;
                int c   = idx >> 5;
                int j16 = (idx & 31) << 2;
                ASYNC_COPY_B128(gx + c * NN + j16, &XBT[nxt][c][j16]);
            }
            if (tid < CHUNK) sqc[nxt][tid] = SQ[b * NN + c0 + CHUNK + tid];
        }

        // ---- G(2x2 tiles) = X_rows . X_colsT over C=64, k=4 f32 WMMAs ----
        const float (*XB)[BPITCH] = XBT[cur];
        v8f acc00 = {}, acc01 = {}, acc10 = {}, acc11 = {};
#pragma unroll
        for (int s = 0; s < 16; ++s) {
            const int cA = 4 * s + ch;
            v2f a0, a1, bv0, bv1;
            a0.x  = XA[m0][cA];   a0.y  = XA[m0][cA + 1];
            a1.x  = XA[m1][cA];   a1.y  = XA[m1][cA + 1];
            bv0.x = XB[cA][n0c];  bv0.y = XB[cA + 1][n0c];
            bv1.x = XB[cA][n1c];  bv1.y = XB[cA + 1][n1c];
            acc00 = __builtin_amdgcn_wmma_f32_16x16x4_f32(false, a0, false, bv0, (short)0, acc00, false, false);
            acc01 = __builtin_amdgcn_wmma_f32_16x16x4_f32(false, a0, false, bv1, (short)0, acc01, false, false);
            acc10 = __builtin_amdgcn_wmma_f32_16x16x4_f32(false, a1, false, bv0, (short)0, acc10, false, false);
            acc11 = __builtin_amdgcn_wmma_f32_16x16x4_f32(false, a1, false, bv1, (short)0, acc11, false, false);
        }

        // dist = |xm|^2 + |xn|^2 - 2 G.  C/D layout: vgpr r holds M=r (lanes0-15)
        // and M=r+8 (lanes16-31); N = lane%16 within the tile.
        const int   rb0 = (2 * rtp) * 16 + 8 * (lane >> 4);
        const int   rb1 = rb0 + 16;
        const int   cb0 = (2 * ctp) * 16 + l16;
        const int   cb1 = cb0 + 16;
        const float sn0 = sqc[cur][cb0];
        const float sn1 = sqc[cur][cb1];
#pragma unroll
        for (int r = 0; r < 8; ++r) {
            int mlA = rb0 + r, mlB = rb1 + r;
            ST[mlA][cb0] = sqrow[mlA] + sn0 - 2.0f * acc00[r];
            ST[mlA][cb1] = sqrow[mlA] + sn1 - 2.0f * acc01[r];
            ST[mlB][cb0] = sqrow[mlB] + sn0 - 2.0f * acc10[r];
            ST[mlB][cb1] = sqrow[mlB] + sn1 - 2.0f * acc11[r];
        }
        __syncthreads();

        // streaming top-16 (self excluded by index; strict '<' reproduces
        // top_k's stable lower-index tie-breaking for ascending scan order).
        // Runs while next chunk's async DMA is in flight.
        if (tid < ROWS) {
            const int m  = tid;
            const int ng = row0 + m;
            for (int j = 0; j < CHUNK; ++j) {
                int n = c0 + j;
                if (n == ng) continue;
                float d = ST[m][j];
                if (d < sel_d[m][KNN - 1]) {
                    int p = KNN - 1;
                    while (p > 0 && sel_d[m][p - 1] > d) {
                        sel_d[m][p] = sel_d[m][p - 1];
                        sel_i[m][p] = sel_i[m][p - 1];
                        --p;
                    }
                    sel_d[m][p] = d;
                    sel_i[m][p] = n;
                }
            }
        }
        __syncthreads();
    }

    // Phase 2: relu(max) aggregation. 64 lanes (2 waves) -> 64 consecutive n
    // per output channel: coalesced stores of out[b][o][row0 .. row0+63].
    for (int pass = 0; pass < 16; ++pass) {
        const int o  = pass * 4 + (tid >> 6);
        const int nl = tid & 63;
        const int ng = row0 + nl;
        const float rv = R[(b * NN + ng) * CO + o];
        float best = -FLT_MAX;
        for (int k = 0; k < KNN; ++k) {
            int m = sel_i[nl][k];
            best = fmaxf(best, rv + P[(b * NN + m) * CO + o]);
        }
        best = fmaxf(best, 0.0f);                // relu commutes with max
        out[(b * CO + o) * NN + ng] = best;
    }
}

// ---------------------------------------------------------------------------
extern "C" void kernel_launch(void* const* d_in, const int* in_sizes, int n_in,
                              void* d_out, int out_size, void* d_ws, size_t ws_size,
                              hipStream_t stream) {
    const float* x  = (const float*)d_in[0];
    const float* W1 = (const float*)d_in[1];
    const float* b1 = (const float*)d_in[2];
    const float* W2 = (const float*)d_in[3];
    const float* b2 = (const float*)d_in[4];
    float* out = (float*)d_out;

    // workspace layout: P [B*N*CO] | R [B*N*CO] | SQ [B*N]  (~16.1 MB)
    float* P  = (float*)d_ws;
    float* R  = P + (size_t)BB * NN * CO;
    float* SQ = R + (size_t)BB * NN * CO;

    edgeconv_precompute<<<BB * (NN / 64), 256, 0, stream>>>(x, W1, b1, W2, b2, P, R, SQ);
    edgeconv_knn_aggregate<<<BB * (NN / ROWS), 256, 0, stream>>>(x, P, R, SQ, out);
}